// PointConvNet_11115375362800
// MI455X (gfx1250) — compile-verified
//
#include <hip/hip_runtime.h>

typedef float v2f __attribute__((ext_vector_type(2)));
typedef float v8f __attribute__((ext_vector_type(8)));

#define BN_EPS 1e-5f

// ============================================================ kernels

// pointcloud [B,7,N] -> xyz [B,N,3], feat [B,N,4]
__global__ void split_input_kernel(const float* __restrict__ pc, float* __restrict__ xyz,
                                   float* __restrict__ feat, int B, int N) {
  int gid = blockIdx.x * blockDim.x + threadIdx.x;
  if (gid >= B * N) return;
  int b = gid / N, n = gid - b * N;
  const float* base = pc + (size_t)b * 7 * N;
  for (int c = 0; c < 3; ++c) xyz[(size_t)gid * 3 + c] = base[(size_t)c * N + n];
  for (int c = 0; c < 4; ++c) feat[(size_t)gid * 4 + c] = base[(size_t)(3 + c) * N + n];
}

// Gaussian KDE inverse density: inv[b,i] = 1 / (mean_j exp(-d2/(2bw^2)) / (2.5 bw))
__global__ void density_kernel(const float* __restrict__ xyz, float* __restrict__ inv,
                               int B, int N, float bw) {
  int gid = blockIdx.x * blockDim.x + threadIdx.x;
  if (gid >= B * N) return;
  int b = gid / N;
  const float* base = xyz + (size_t)b * N * 3;
  const float* p = xyz + (size_t)gid * 3;
  float px = p[0], py = p[1], pz = p[2];
  float s = -1.0f / (2.0f * bw * bw);
  float acc = 0.f;
  for (int j = 0; j < N; ++j) {
    float dx = base[j * 3 + 0] - px;
    float dy = base[j * 3 + 1] - py;
    float dz = base[j * 3 + 2] - pz;
    acc += expf((dx * dx + dy * dy + dz * dz) * s);
  }
  float dens = (acc / (2.5f * bw)) / (float)N;
  inv[gid] = 1.0f / dens;
}

// Farthest point sampling; one block per batch. Writes new_xyz [B,S,3].
__global__ void fps_kernel(const float* __restrict__ xyz, int N, int npoint,
                           float* __restrict__ newXyz) {
  int b = blockIdx.x;
  const float* p = xyz + (size_t)b * N * 3;
  __shared__ float dist[4096];
  __shared__ float rv[256];
  __shared__ int ri[256];
  __shared__ int far;
  int t = threadIdx.x;
  for (int i = t; i < N; i += 256) dist[i] = 1e10f;
  if (t == 0) far = 0;
  __syncthreads();
  for (int it = 0; it < npoint; ++it) {
    int f = far;
    float cx = p[f * 3 + 0], cy = p[f * 3 + 1], cz = p[f * 3 + 2];
    if (t == 0) {
      newXyz[((size_t)b * npoint + it) * 3 + 0] = cx;
      newXyz[((size_t)b * npoint + it) * 3 + 1] = cy;
      newXyz[((size_t)b * npoint + it) * 3 + 2] = cz;
    }
    float bestv = -1.f;
    int besti = 0;
    for (int i = t; i < N; i += 256) {
      float dx = p[i * 3 + 0] - cx, dy = p[i * 3 + 1] - cy, dz = p[i * 3 + 2] - cz;
      float d = dx * dx + dy * dy + dz * dz;
      float nd = fminf(dist[i], d);
      dist[i] = nd;
      if (nd > bestv) { bestv = nd; besti = i; }
    }
    rv[t] = bestv; ri[t] = besti;
    __syncthreads();
    for (int s = 128; s > 0; s >>= 1) {
      if (t < s) {
        if (rv[t + s] > rv[t] || (rv[t + s] == rv[t] && ri[t + s] < ri[t])) {
          rv[t] = rv[t + s]; ri[t] = ri[t + s];
        }
      }
      __syncthreads();
    }
    if (t == 0) far = ri[0];
    __syncthreads();
  }
}

// kNN (K=32) with stable tie-break (earlier index wins), one thread per query.
__global__ void knn_kernel(const float* __restrict__ q, const float* __restrict__ ref,
                           int* __restrict__ outIdx, int B, int S, int N) {
  int gid = blockIdx.x * blockDim.x + threadIdx.x;
  if (gid >= B * S) return;
  int b = gid / S;
  const float* qq = q + (size_t)gid * 3;
  const float* r = ref + (size_t)b * N * 3;
  float qx = qq[0], qy = qq[1], qz = qq[2];
  float kd[32]; int ki[32];
  for (int k = 0; k < 32; ++k) { kd[k] = 3.4e38f; ki[k] = 0; }
  for (int j = 0; j < N; ++j) {
    float dx = r[j * 3 + 0] - qx, dy = r[j * 3 + 1] - qy, dz = r[j * 3 + 2] - qz;
    float d = dx * dx + dy * dy + dz * dz;
    if (d < kd[31]) {
      int pos = 31;
      while (pos > 0 && kd[pos - 1] > d) { kd[pos] = kd[pos - 1]; ki[pos] = ki[pos - 1]; --pos; }
      kd[pos] = d; ki[pos] = j;
    }
  }
  for (int k = 0; k < 32; ++k) outIdx[(size_t)gid * 32 + k] = ki[k];
}

// Build new_points [BSK, 3+C], grouped_xyz_norm [BSK,3], grouped inv_density [BSK]
__global__ void group_kernel(const float* __restrict__ xyz, const float* __restrict__ newxyz,
                             const float* __restrict__ pts, const float* __restrict__ invden,
                             const int* __restrict__ idx, float* __restrict__ npnts,
                             float* __restrict__ gnorm, float* __restrict__ gden,
                             int B, int S, int N, int C) {
  int gid = blockIdx.x * blockDim.x + threadIdx.x;
  if (gid >= B * S * 32) return;
  int bs = gid >> 5;
  int b = bs / S;
  int j = idx[gid];
  const float* pj = xyz + ((size_t)b * N + j) * 3;
  const float* cc = newxyz + (size_t)bs * 3;
  float nx = pj[0] - cc[0], ny = pj[1] - cc[1], nz = pj[2] - cc[2];
  gnorm[(size_t)gid * 3 + 0] = nx;
  gnorm[(size_t)gid * 3 + 1] = ny;
  gnorm[(size_t)gid * 3 + 2] = nz;
  float* np = npnts + (size_t)gid * (3 + C);
  np[0] = nx; np[1] = ny; np[2] = nz;
  const float* f = pts + ((size_t)b * N + j) * C;
  for (int c = 0; c < C; ++c) np[3 + c] = f[c];
  gden[gid] = invden[(size_t)b * N + j];
}

// DensityNet input normalization: per (b,s) divide by max over K
__global__ void dnorm_kernel(const float* __restrict__ g, float* __restrict__ o, int BS) {
  int gid = blockIdx.x * blockDim.x + threadIdx.x;
  if (gid >= BS) return;
  const float* a = g + (size_t)gid * 32;
  float m = a[0];
  for (int k = 1; k < 32; ++k) m = fmaxf(m, a[k]);
  for (int k = 0; k < 32; ++k) o[(size_t)gid * 32 + k] = a[k] / m;
}

// h[m, c] *= d[m]
__global__ void scale_kernel(float* __restrict__ h, const float* __restrict__ d,
                             size_t total, int C) {
  size_t idx = (size_t)blockIdx.x * blockDim.x + threadIdx.x;
  if (idx >= total) return;
  h[idx] *= d[idx / (size_t)C];
}

// Fused GEMM: Y = act( (X[M,K] @ W[Cout,K]^T + cb) * bn_scale + bn_shift )
// flags bit0: BN, bit1: ReLU. f32 WMMA 16x16x4; one wave computes a 16 x (16*NT)
// tile (A fragment reused across NT B fragments). Inner loop is branch-free:
// the aligned body does unconditional loads (vectorized b64 when K is even);
// the K%4 tail uses clamped addresses + selects (v_cndmask, no EXEC traffic).
template <int NT>
__global__ void gemm_bn_act_kernel(const float* __restrict__ X, const float* __restrict__ W,
                                   const float* __restrict__ cb,
                                   const float* __restrict__ bg, const float* __restrict__ bb,
                                   const float* __restrict__ bm, const float* __restrict__ bv,
                                   float* __restrict__ Y, int M, int K, int Cout, int flags) {
  int wave = (blockIdx.x * blockDim.x + threadIdx.x) >> 5;
  int lane = threadIdx.x & 31;
  int ntg = (Cout + 16 * NT - 1) / (16 * NT);
  int mtiles = (M + 15) >> 4;
  int mt = wave / ntg;
  int ng = wave - mt * ntg;
  if (mt >= mtiles) return;  // wave-uniform exit; EXEC stays full around WMMA
  int m0 = mt * 16, n0 = ng * 16 * NT;
  int lrow = lane & 15;
  int hi = lane >> 4;        // 0 or 1
  int khalf = hi * 2;        // lanes 0-15 carry K+0/K+1, lanes 16-31 carry K+2/K+3
  int arow = m0 + lrow; if (arow > M - 1) arow = M - 1;
  const float* Xrow = X + (size_t)arow * K;
  const float* Wrow[NT];
#pragma unroll
  for (int j = 0; j < NT; ++j) {
    int bcol = n0 + j * 16 + lrow;
    if (bcol > Cout - 1) bcol = Cout - 1;
    Wrow[j] = W + (size_t)bcol * K;
  }
  v8f acc[NT];
#pragma unroll
  for (int j = 0; j < NT; ++j) acc[j] = (v8f){};

  int Ka = K & ~3;
  if (!(K & 1)) {
    // K even: arow*K and k+khalf are even -> 8B-aligned v2f loads
    const v2f* Xv = (const v2f*)Xrow;
    const v2f* Wv[NT];
#pragma unroll
    for (int j = 0; j < NT; ++j) Wv[j] = (const v2f*)Wrow[j];
    for (int k = 0; k < Ka; k += 4) {
      v2f a = Xv[(k >> 1) + hi];
#pragma unroll
      for (int j = 0; j < NT; ++j) {
        v2f b = Wv[j][(k >> 1) + hi];
        acc[j] = __builtin_amdgcn_wmma_f32_16x16x4_f32(false, a, false, b, (short)0,
                                                       acc[j], false, false);
      }
    }
  } else {
    for (int k = 0; k < Ka; k += 4) {
      v2f a;
      a.x = Xrow[k + khalf];
      a.y = Xrow[k + khalf + 1];
#pragma unroll
      for (int j = 0; j < NT; ++j) {
        v2f b;
        b.x = Wrow[j][k + khalf];
        b.y = Wrow[j][k + khalf + 1];
        acc[j] = __builtin_amdgcn_wmma_f32_16x16x4_f32(false, a, false, b, (short)0,
                                                       acc[j], false, false);
      }
    }
  }
  if (Ka < K) {
    // tail: clamped in-bounds loads + selects (no divergent loads)
    int k0 = Ka + khalf, k1 = Ka + khalf + 1;
    int c0 = (k0 < K) ? k0 : (K - 1);
    int c1 = (k1 < K) ? k1 : (K - 1);
    v2f a;
    float ax = Xrow[c0], ay = Xrow[c1];
    a.x = (k0 < K) ? ax : 0.f;
    a.y = (k1 < K) ? ay : 0.f;
#pragma unroll
    for (int j = 0; j < NT; ++j) {
      v2f b;
      float bx = Wrow[j][c0], by = Wrow[j][c1];
      b.x = (k0 < K) ? bx : 0.f;
      b.y = (k1 < K) ? by : 0.f;
      acc[j] = __builtin_amdgcn_wmma_f32_16x16x4_f32(false, a, false, b, (short)0,
                                                     acc[j], false, false);
    }
  }

  // epilogue: bias + folded BN + optional ReLU, masked stores
#pragma unroll
  for (int j = 0; j < NT; ++j) {
    int col = n0 + j * 16 + lrow;
    if (col < Cout) {
      float bias = cb ? cb[col] : 0.f;
      float s = 1.f, t = 0.f;
      if (flags & 1) {
        float invs = rsqrtf(bv[col] + BN_EPS);
        s = bg[col] * invs;
        t = bb[col] - bm[col] * s;
      }
#pragma unroll
      for (int r = 0; r < 8; ++r) {
        int row = m0 + r + hi * 8;
        if (row < M) {
          float val = (acc[j][r] + bias) * s + t;
          if (flags & 2) val = fmaxf(val, 0.f);
          Y[(size_t)row * Cout + col] = val;
        }
      }
    }
  }
}

// agg[bs, c, w] = sum_k x[bs,k,c] * wgt[bs,k,w] ; M=C tile, N=16, K=32 per (b,s)
__global__ void einsum_wmma_kernel(const float* __restrict__ Xf, const float* __restrict__ Wf,
                                   float* __restrict__ Agg, int BS, int C) {
  int wave = (blockIdx.x * blockDim.x + threadIdx.x) >> 5;
  int lane = threadIdx.x & 31;
  int ctiles = C >> 4;
  int bs = wave / ctiles;
  int ct = wave - bs * ctiles;
  if (bs >= BS) return;
  const float* x = Xf + (size_t)bs * 32 * C;
  const float* w = Wf + (size_t)bs * 32 * 16;
  int lrow = lane & 15;
  int hi = lane >> 4;
  int khalf = hi * 2;
  int c0 = ct * 16;
  v8f acc = {};
  for (int k = 0; k < 32; k += 4) {
    v2f a, b;
    a.x = x[(size_t)(k + khalf) * C + c0 + lrow];       // A[m,k] = x[k, c0+m]
    a.y = x[(size_t)(k + khalf + 1) * C + c0 + lrow];
    b.x = w[(size_t)(k + khalf) * 16 + lrow];
    b.y = w[(size_t)(k + khalf + 1) * 16 + lrow];
    acc = __builtin_amdgcn_wmma_f32_16x16x4_f32(false, a, false, b, (short)0, acc,
                                                false, false);
  }
#pragma unroll
  for (int r = 0; r < 8; ++r) {
    int row = c0 + r + hi * 8;
    Agg[((size_t)bs * C + row) * 16 + lrow] = acc[r];
  }
}

// 3-NN with inverse-distance weights (normalized)
__global__ void three_nn_kernel(const float* __restrict__ q, const float* __restrict__ ref,
                                int* __restrict__ oi, float* __restrict__ ow,
                                int B, int N, int S) {
  int gid = blockIdx.x * blockDim.x + threadIdx.x;
  if (gid >= B * N) return;
  int b = gid / N;
  const float* qq = q + (size_t)gid * 3;
  const float* r = ref + (size_t)b * S * 3;
  float qx = qq[0], qy = qq[1], qz = qq[2];
  float kd[3] = {3.4e38f, 3.4e38f, 3.4e38f};
  int ki[3] = {0, 0, 0};
  for (int j = 0; j < S; ++j) {
    float dx = r[j * 3 + 0] - qx, dy = r[j * 3 + 1] - qy, dz = r[j * 3 + 2] - qz;
    float d = dx * dx + dy * dy + dz * dz;
    if (d < kd[2]) {
      int pos = 2;
      while (pos > 0 && kd[pos - 1] > d) { kd[pos] = kd[pos - 1]; ki[pos] = ki[pos - 1]; --pos; }
      kd[pos] = d; ki[pos] = j;
    }
  }
  float w0 = 1.f / (kd[0] + 1e-8f), w1 = 1.f / (kd[1] + 1e-8f), w2 = 1.f / (kd[2] + 1e-8f);
  float ws = w0 + w1 + w2;
  oi[(size_t)gid * 3 + 0] = ki[0];
  oi[(size_t)gid * 3 + 1] = ki[1];
  oi[(size_t)gid * 3 + 2] = ki[2];
  ow[(size_t)gid * 3 + 0] = w0 / ws;
  ow[(size_t)gid * 3 + 1] = w1 / ws;
  ow[(size_t)gid * 3 + 2] = w2 / ws;
}

// X[b,n,:] = concat(p1[b,n,:C1], sum_j w_j * p2[b, idx_j, :C2])
__global__ void interp_concat_kernel(const float* __restrict__ p1, int C1,
                                     const float* __restrict__ p2, int C2,
                                     const int* __restrict__ oi, const float* __restrict__ ow,
                                     float* __restrict__ X, int B, int N, int S) {
  int gid = blockIdx.x * blockDim.x + threadIdx.x;
  if (gid >= B * N) return;
  int b = gid / N;
  float* o = X + (size_t)gid * (C1 + C2);
  if (p1) {
    const float* a = p1 + (size_t)gid * C1;
    for (int c = 0; c < C1; ++c) o[c] = a[c];
  }
  const int* id = oi + (size_t)gid * 3;
  const float* w = ow + (size_t)gid * 3;
  const float* base = p2 + (size_t)b * S * C2;
  const float* r0 = base + (size_t)id[0] * C2;
  const float* r1 = base + (size_t)id[1] * C2;
  const float* r2 = base + (size_t)id[2] * C2;
  float w0 = w[0], w1 = w[1], w2 = w[2];
  for (int d = 0; d < C2; ++d) o[C1 + d] = r0[d] * w0 + r1[d] * w1 + r2[d] * w2;
}

// logits [B*N,8] -> log_softmax transposed to [B,8,N]
__global__ void logsoftmax_kernel(const float* __restrict__ lg, float* __restrict__ out,
                                  int B, int N) {
  int gid = blockIdx.x * blockDim.x + threadIdx.x;
  if (gid >= B * N) return;
  int b = gid / N, n = gid - b * N;
  const float* x = lg + (size_t)gid * 8;
  float m = x[0];
  for (int c = 1; c < 8; ++c) m = fmaxf(m, x[c]);
  float s = 0.f;
  for (int c = 0; c < 8; ++c) s += expf(x[c] - m);
  float lse = m + logf(s);
  for (int c = 0; c < 8; ++c) out[((size_t)b * 8 + c) * N + n] = x[c] - lse;
}

// ============================================================ host side

namespace {

struct Conv { const float* w; const float* b; int cin; int cout; };
struct Bn { const float* g; const float* b; const float* m; const float* v; };
struct Layer { Conv c; Bn n; };
struct Enc { Layer mlp[2]; Layer wn[3]; Layer dn[3]; Conv lin; Bn bnlin; };
struct Params {
  Enc sa[3];
  Layer fp3[2], fp2[2], fp1[2];
  Conv conv1; Bn bn1; Conv conv2;
};

struct Reader {
  void* const* din; const int* sz; int pos; int n; bool ok; bool srt;
  const float* take(int need) {
    if (pos >= n || sz[pos] != need) { ok = false; return nullptr; }
    return (const float*)din[pos++];
  }
};

static Conv rdConv(Reader& r, int ci, int co) {
  Conv c; c.cin = ci; c.cout = co;
  if (r.srt) { c.b = r.take(co); c.w = r.take(ci * co); }
  else       { c.w = r.take(ci * co); c.b = r.take(co); }
  return c;
}
static Bn rdBn(Reader& r, int c) {
  Bn b;
  if (r.srt) { b.b = r.take(c); b.g = r.take(c); b.m = r.take(c); b.v = r.take(c); }
  else       { b.g = r.take(c); b.b = r.take(c); b.m = r.take(c); b.v = r.take(c); }
  return b;
}
static void rdChain(Reader& r, Layer* L, const int* dims, int nl) {
  for (int i = 0; i < nl; ++i) { L[i].c = rdConv(r, dims[i], dims[i + 1]); L[i].n = rdBn(r, dims[i + 1]); }
}
static Enc rdEnc(Reader& r, int cin, int m0, int m1) {
  Enc e;
  int md[3] = {cin, m0, m1};
  int wd[4] = {3, 8, 8, 16};
  int dd[4] = {1, 16, 8, 1};
  if (r.srt) {  // sorted keys: bn_lin, dn, lin, mlp, wn
    e.bnlin = rdBn(r, m1);
    rdChain(r, e.dn, dd, 3);
    e.lin = rdConv(r, 16 * m1, m1);
    rdChain(r, e.mlp, md, 2);
    rdChain(r, e.wn, wd, 3);
  } else {      // insertion: mlp, wn, dn, lin, bn_lin
    rdChain(r, e.mlp, md, 2);
    rdChain(r, e.wn, wd, 3);
    rdChain(r, e.dn, dd, 3);
    e.lin = rdConv(r, 16 * m1, m1);
    e.bnlin = rdBn(r, m1);
  }
  return e;
}
static bool rdParams(Reader& r, Params& P) {
  int f3[3] = {384, 256, 256}, f2[3] = {320, 256, 256}, f1[3] = {256, 128, 128};
  if (r.srt) {  // bn1, conv1, conv2, fp1, fp2, fp3, sa1, sa2, sa3
    P.bn1 = rdBn(r, 128);
    P.conv1 = rdConv(r, 128, 128);
    P.conv2 = rdConv(r, 128, 8);
    rdChain(r, P.fp1, f1, 2);
    rdChain(r, P.fp2, f2, 2);
    rdChain(r, P.fp3, f3, 2);
    P.sa[0] = rdEnc(r, 7, 32, 64);
    P.sa[1] = rdEnc(r, 67, 64, 128);
    P.sa[2] = rdEnc(r, 131, 128, 256);
  } else {      // sa1, sa2, sa3, fp3, fp2, fp1, conv1, bn1, conv2
    P.sa[0] = rdEnc(r, 7, 32, 64);
    P.sa[1] = rdEnc(r, 67, 64, 128);
    P.sa[2] = rdEnc(r, 131, 128, 256);
    rdChain(r, P.fp3, f3, 2);
    rdChain(r, P.fp2, f2, 2);
    rdChain(r, P.fp1, f1, 2);
    P.conv1 = rdConv(r, 128, 128);
    P.bn1 = rdBn(r, 128);
    P.conv2 = rdConv(r, 128, 8);
  }
  return r.ok;
}

struct WS {
  char* base; size_t cap; size_t off; bool ok;
  float* f(size_t n) {
    size_t bytes = (n * 4 + 255) & ~(size_t)255;
    if (off + bytes > cap) { ok = false; return (float*)base; }
    float* p = (float*)(base + off); off += bytes; return p;
  }
  int* i(size_t n) { return (int*)f(n); }
};

static const int B = 4;

static void gemm(hipStream_t st, const float* X, const Conv& c, const Bn* bn,
                 float* Y, int M, int flags) {
  int mt = (M + 15) / 16;
  const float* g = bn ? bn->g : nullptr;
  const float* bb = bn ? bn->b : nullptr;
  const float* bm = bn ? bn->m : nullptr;
  const float* bv = bn ? bn->v : nullptr;
  if (c.cout >= 32) {
    int ntg = (c.cout + 31) / 32;
    int waves = mt * ntg;
    gemm_bn_act_kernel<2><<<dim3((waves + 3) / 4), dim3(128), 0, st>>>(
        X, c.w, c.b, g, bb, bm, bv, Y, M, c.cin, c.cout, flags);
  } else {
    int ntg = (c.cout + 15) / 16;
    int waves = mt * ntg;
    gemm_bn_act_kernel<1><<<dim3((waves + 3) / 4), dim3(128), 0, st>>>(
        X, c.w, c.b, g, bb, bm, bv, Y, M, c.cin, c.cout, flags);
  }
}

static void run_encoder(hipStream_t st, WS& ws, const float* xyz, const float* pts,
                        int Nin, int Cpts, int S, float bw, const Enc& e, int m0, int m1,
                        float* newXyz, float* outP) {
  size_t mark = ws.off;
  int BS = B * S, BSK = BS * 32;
  int cinAll = 3 + Cpts;
  float* invden = ws.f((size_t)B * Nin);
  int* knnIdx = ws.i((size_t)BSK);
  float* npnts = ws.f((size_t)BSK * cinAll);
  float* gnorm = ws.f((size_t)BSK * 3);
  float* gden  = ws.f((size_t)BSK);
  float* h1    = ws.f((size_t)BSK * m0);
  float* h2    = ws.f((size_t)BSK * m1);
  float* dnin  = ws.f((size_t)BSK);
  float* d16   = ws.f((size_t)BSK * 16);
  float* d8    = ws.f((size_t)BSK * 8);
  float* d1    = ws.f((size_t)BSK);
  float* w8a   = ws.f((size_t)BSK * 8);
  float* w8b   = ws.f((size_t)BSK * 8);
  float* w16   = ws.f((size_t)BSK * 16);
  float* agg   = ws.f((size_t)BS * m1 * 16);
  if (!ws.ok) return;

  int gBN = (B * Nin + 255) / 256;
  int gBS = (BS + 255) / 256;
  int gBSK = (BSK + 255) / 256;

  density_kernel<<<gBN, 256, 0, st>>>(xyz, invden, B, Nin, bw);
  fps_kernel<<<B, 256, 0, st>>>(xyz, Nin, S, newXyz);
  knn_kernel<<<gBS, 256, 0, st>>>(newXyz, xyz, knnIdx, B, S, Nin);
  group_kernel<<<gBSK, 256, 0, st>>>(xyz, newXyz, pts, invden, knnIdx,
                                     npnts, gnorm, gden, B, S, Nin, Cpts);
  // main MLP chain
  gemm(st, npnts, e.mlp[0].c, &e.mlp[0].n, h1, BSK, 3);
  gemm(st, h1, e.mlp[1].c, &e.mlp[1].n, h2, BSK, 3);
  // DensityNet
  dnorm_kernel<<<gBS, 256, 0, st>>>(gden, dnin, BS);
  gemm(st, dnin, e.dn[0].c, &e.dn[0].n, d16, BSK, 3);
  gemm(st, d16, e.dn[1].c, &e.dn[1].n, d8, BSK, 3);
  gemm(st, d8, e.dn[2].c, &e.dn[2].n, d1, BSK, 3);
  size_t tot = (size_t)BSK * m1;
  scale_kernel<<<(unsigned)((tot + 255) / 256), 256, 0, st>>>(h2, d1, tot, m1);
  // WeightNet
  gemm(st, gnorm, e.wn[0].c, &e.wn[0].n, w8a, BSK, 3);
  gemm(st, w8a, e.wn[1].c, &e.wn[1].n, w8b, BSK, 3);
  gemm(st, w8b, e.wn[2].c, &e.wn[2].n, w16, BSK, 3);
  // einsum bskc,bskw->bscw
  {
    int waves = BS * (m1 / 16);
    einsum_wmma_kernel<<<(waves + 3) / 4, 128, 0, st>>>(h2, w16, agg, BS, m1);
  }
  gemm(st, agg, e.lin, &e.bnlin, outP, BS, 3);
  ws.off = mark;
}

static void run_decoder(hipStream_t st, WS& ws, const float* xyz1, int N1,
                        const float* xyz2, int S2, const float* p1, int C1,
                        const float* p2, int C2, const Layer* mlp, float* out) {
  size_t mark = ws.off;
  int* oi = ws.i((size_t)B * N1 * 3);
  float* ow = ws.f((size_t)B * N1 * 3);
  float* X = ws.f((size_t)B * N1 * (C1 + C2));
  float* mid = ws.f((size_t)B * N1 * mlp[0].c.cout);
  if (!ws.ok) return;
  int gBN = (B * N1 + 255) / 256;
  three_nn_kernel<<<gBN, 256, 0, st>>>(xyz1, xyz2, oi, ow, B, N1, S2);
  interp_concat_kernel<<<gBN, 256, 0, st>>>(p1, C1, p2, C2, oi, ow, X, B, N1, S2);
  gemm(st, X, mlp[0].c, &mlp[0].n, mid, B * N1, 3);
  gemm(st, mid, mlp[1].c, &mlp[1].n, out, B * N1, 3);
  ws.off = mark;
}

}  // namespace

extern "C" void kernel_launch(void* const* d_in, const int* in_sizes, int n_in,
                              void* d_out, int out_size, void* d_ws, size_t ws_size,
                              hipStream_t stream) {
  const int N0 = 4096;
  const int PC_SIZE = B * 7 * N0;

  // locate pointcloud and params (handles both flatten orders)
  int pcIdx = -1, pstart = 0;
  if (n_in > 0 && in_sizes[0] == PC_SIZE) { pcIdx = 0; pstart = 1; }
  else if (n_in > 0 && in_sizes[n_in - 1] == PC_SIZE) { pcIdx = n_in - 1; pstart = 0; }
  else return;

  Params P;
  bool got = false;
  for (int s = 0; s < 2 && !got; ++s) {
    Reader r{d_in, in_sizes, pstart, n_in, true, s == 1};
    Params tmp;
    if (rdParams(r, tmp)) { P = tmp; got = true; }
  }
  if (!got) return;
  const float* pc = (const float*)d_in[pcIdx];
  float* dout = (float*)d_out;

  WS ws{(char*)d_ws, ws_size, 0, true};
  // persistent buffers
  float* xyz0 = ws.f((size_t)B * N0 * 3);
  float* feat0 = ws.f((size_t)B * N0 * 4);
  float* l1x = ws.f((size_t)B * 1024 * 3);
  float* l1p = ws.f((size_t)B * 1024 * 64);
  float* l2x = ws.f((size_t)B * 256 * 3);
  float* l2p = ws.f((size_t)B * 256 * 128);
  float* l3x = ws.f((size_t)B * 64 * 3);
  float* l3p = ws.f((size_t)B * 64 * 256);
  float* d2p = ws.f((size_t)B * 256 * 256);
  float* d1p = ws.f((size_t)B * 1024 * 256);
  float* l0p = ws.f((size_t)B * N0 * 128);
  float* hbuf = ws.f((size_t)B * N0 * 128);
  float* logits = ws.f((size_t)B * N0 * 8);
  if (!ws.ok) return;

  split_input_kernel<<<(B * N0 + 255) / 256, 256, 0, stream>>>(pc, xyz0, feat0, B, N0);

  run_encoder(stream, ws, xyz0, feat0, N0, 4, 1024, 0.1f, P.sa[0], 32, 64, l1x, l1p);
  run_encoder(stream, ws, l1x, l1p, 1024, 64, 256, 0.2f, P.sa[1], 64, 128, l2x, l2p);
  run_encoder(stream, ws, l2x, l2p, 256, 128, 64, 0.4f, P.sa[2], 128, 256, l3x, l3p);

  run_decoder(stream, ws, l2x, 256, l3x, 64, l2p, 128, l3p, 256, P.fp3, d2p);
  run_decoder(stream, ws, l1x, 1024, l2x, 256, l1p, 64, d2p, 256, P.fp2, d1p);
  run_decoder(stream, ws, xyz0, N0, l1x, 1024, nullptr, 0, d1p, 256, P.fp1, l0p);

  gemm(stream, l0p, P.conv1, &P.bn1, hbuf, B * N0, 3);
  gemm(stream, hbuf, P.conv2, nullptr, logits, B * N0, 0);
  logsoftmax_kernel<<<(B * N0 + 255) / 256, 256, 0, stream>>>(logits, dout, B, N0);
}